// Gen_67456756351234
// MI455X (gfx1250) — compile-verified
//
#include <hip/hip_runtime.h>
#include <hip/hip_bf16.h>

typedef float v2f __attribute__((ext_vector_type(2)));
typedef float v8f __attribute__((ext_vector_type(8)));

#define NU 100000
#define NI 50000
#define GD 64
#define NE 3200000
#define KD 128
#define DEPTH 3
#define CMIX 0.9f
#define EPSV 1e-16f

// Native f32 atomic add (agent scope, relaxed) -> global_atomic_add_f32, no CAS loop.
__device__ __forceinline__ void atomAddF(float* p, float v) {
  __hip_atomic_fetch_add(p, v, __ATOMIC_RELAXED, __HIP_MEMORY_SCOPE_AGENT);
}

__global__ void k_zero(float* __restrict__ p, int n) {
  int i = blockIdx.x * blockDim.x + threadIdx.x;
  int st = gridDim.x * blockDim.x;
  for (; i < n; i += st) p[i] = 0.0f;
}

// expla1 -> out_lfla1 slot (pre-normalization); sumw1[u] = eps + rowsum(expla1)
__global__ void k_user_init(const float* __restrict__ la1,
                            float* __restrict__ expla1,
                            float* __restrict__ sumw1) {
  int u = blockIdx.x * blockDim.x + threadIdx.x;
  if (u >= NU) return;
  float s = EPSV;
#pragma unroll 8
  for (int d = 0; d < GD; ++d) {
    float e = expf(la1[u * GD + d]);
    expla1[u * GD + d] = e;
    s += e;
  }
  sumw1[u] = s;
}

__global__ void k_small_init(float* __restrict__ sumw2,
                             float* __restrict__ fa0,
                             float* __restrict__ col2) {
  int i = blockIdx.x * blockDim.x + threadIdx.x;
  if (i < NI) sumw2[i] = EPSV;
  if (i < NU) fa0[i] = 0.0f;
  if (i < GD) col2[i] = 0.0f;
}

// expla2 -> out_lfla2 slot; column sums of expla2 into col2[64]
__global__ void k_la2_exp_colsum(const float* __restrict__ la2,
                                 float* __restrict__ expla2,
                                 float* __restrict__ col2) {
  __shared__ float sm[256];
  int t = threadIdx.x;
  int col = t & (GD - 1);
  int r0 = t >> 6;  // 0..3
  float local = 0.0f;
  for (int u = blockIdx.x * 4 + r0; u < NU; u += gridDim.x * 4) {
    float e = expf(la2[u * GD + col]);
    expla2[u * GD + col] = e;
    local += e;
  }
  sm[t] = local;
  __syncthreads();
  if (t < GD) atomAddF(col2 + t, sm[t] + sm[t + 64] + sm[t + 128] + sm[t + 192]);
}

// exp(w1/w2) stored into lfw slots; segment-sum into sumw1/sumw2
__global__ void k_edge_exp(const float* __restrict__ w1, const float* __restrict__ w2,
                           const int* __restrict__ r1, const int* __restrict__ r2,
                           float* __restrict__ ew1, float* __restrict__ ew2,
                           float* __restrict__ sumw1, float* __restrict__ sumw2) {
  int i = blockIdx.x * blockDim.x + threadIdx.x;
  int st = gridDim.x * blockDim.x;
  for (; i < NE; i += st) {
    float a = expf(w1[i]);
    float b = expf(w2[i]);
    ew1[i] = a;
    ew2[i] = b;
    atomAddF(sumw1 + r1[i], a);
    atomAddF(sumw2 + r2[i], b);
  }
}

// lfw = exp / sum ; fa0[u] += lfw1*lfw2
__global__ void k_edge_norm(const int* __restrict__ r1, const int* __restrict__ r2,
                            float* __restrict__ lfw1, float* __restrict__ lfw2,
                            const float* __restrict__ sumw1, const float* __restrict__ sumw2,
                            float* __restrict__ fa0) {
  int i = blockIdx.x * blockDim.x + threadIdx.x;
  int st = gridDim.x * blockDim.x;
  for (; i < NE; i += st) {
    float a = lfw1[i] / sumw1[r1[i]];
    float b = lfw2[i] / sumw2[r2[i]];
    lfw1[i] = a;
    lfw2[i] = b;
    atomAddF(fa0 + r1[i], a * b);
  }
}

__global__ void k_la_norm(float* __restrict__ lfla1, float* __restrict__ lfla2,
                          const float* __restrict__ sumw1, const float* __restrict__ col2) {
  int i = blockIdx.x * blockDim.x + threadIdx.x;
  if (i >= NU * GD) return;
  int u = i >> 6;
  int d = i & (GD - 1);
  lfla1[i] = lfla1[i] / sumw1[u];
  lfla2[i] = lfla2[i] / (col2[d] + EPSV);
}

// column sums of f (for the per-step mean)
__global__ void k_colsum_f(const float* __restrict__ f, float* __restrict__ sumf) {
  __shared__ float sm[256];
  int t = threadIdx.x;
  int col = t & (KD - 1);
  int r0 = t >> 7;  // 0..1
  float local = 0.0f;
  for (int u = blockIdx.x * 2 + r0; u < NU; u += gridDim.x * 2) local += f[u * KD + col];
  sm[t] = local;
  __syncthreads();
  if (t < KD) atomAddF(sumf + t, sm[t] + sm[t + 128]);
}

// y[r2[e],:] += lfw2[e] * f[r1[e],:]   (one wave per edge, float4 per lane)
__global__ void k_scatter_y(const float* __restrict__ f, const float* __restrict__ lfw2,
                            const int* __restrict__ r1, const int* __restrict__ r2,
                            float* __restrict__ y) {
  int lane = threadIdx.x & 31;
  int wv = (blockIdx.x * blockDim.x + threadIdx.x) >> 5;
  int nw = (gridDim.x * blockDim.x) >> 5;
  for (int e = wv; e < NE; e += nw) {
    float w = lfw2[e];
    int u = r1[e];
    int it = r2[e];
    float4 v = *(const float4*)(f + (size_t)u * KD + lane * 4);
    float* yp = y + (size_t)it * KD + lane * 4;
    atomAddF(yp + 0, w * v.x);
    atomAddF(yp + 1, w * v.y);
    atomAddF(yp + 2, w * v.z);
    atomAddF(yp + 3, w * v.w);
  }
}

// z[r1[e],:] += lfw1[e] * y[r2[e],:]
__global__ void k_scatter_z(const float* __restrict__ y, const float* __restrict__ lfw1,
                            const int* __restrict__ r1, const int* __restrict__ r2,
                            float* __restrict__ z) {
  int lane = threadIdx.x & 31;
  int wv = (blockIdx.x * blockDim.x + threadIdx.x) >> 5;
  int nw = (gridDim.x * blockDim.x) >> 5;
  for (int e = wv; e < NE; e += nw) {
    float w = lfw1[e];
    int u = r1[e];
    int it = r2[e];
    float4 v = *(const float4*)(y + (size_t)it * KD + lane * 4);
    float* zp = z + (size_t)u * KD + lane * 4;
    atomAddF(zp + 0, w * v.x);
    atomAddF(zp + 1, w * v.y);
    atomAddF(zp + 2, w * v.z);
    atomAddF(zp + 3, w * v.w);
  }
}

// t[64,128] = lfla2^T @ f   -- split-K over users, f32 WMMA 16x16x4, atomic accumulate
__global__ void k_wmma_t(const float* __restrict__ lfla2, const float* __restrict__ f,
                         float* __restrict__ t) {
  int lane = threadIdx.x & 31;
  int wv = threadIdx.x >> 5;  // 0..7: each wave owns 4 of the 4x8 output tiles
  int l15 = lane & 15;
  int hi = lane >> 4;  // 0 or 1 -> K offset 2*hi
  int per = (NU + gridDim.x - 1) / gridDim.x;
  per = (per + 3) & ~3;
  int ub = blockIdx.x * per;
  int ue = min(ub + per, NU);
  v8f acc[4] = {};
  for (int u = ub; u < ue; u += 4) {
#pragma unroll
    for (int j = 0; j < 4; ++j) {
      int tile = wv * 4 + j;
      int td = tile >> 3;  // d-tile 0..3
      int tk = tile & 7;   // k-tile 0..7
      int d = td * 16 + l15;
      v2f a, b;
      // A[M=d, K=user] = lfla2[user, d]; lane hi selects K=2*hi+{0,1}
      a.x = lfla2[(u + 2 * hi + 0) * GD + d];
      a.y = lfla2[(u + 2 * hi + 1) * GD + d];
      // B[K=user, N=kcol] = f[user, kcol]
      b.x = f[(u + 2 * hi + 0) * KD + tk * 16 + l15];
      b.y = f[(u + 2 * hi + 1) * KD + tk * 16 + l15];
      acc[j] = __builtin_amdgcn_wmma_f32_16x16x4_f32(false, a, false, b, (short)0, acc[j],
                                                     false, false);
    }
  }
#pragma unroll
  for (int j = 0; j < 4; ++j) {
    int tile = wv * 4 + j;
    int td = tile >> 3;
    int tk = tile & 7;
#pragma unroll
    for (int r = 0; r < 8; ++r) {
      int m = td * 16 + hi * 8 + r;   // C layout: VGPR r, lanes16-31 are M+8
      int n = tk * 16 + l15;
      atomAddF(t + m * KD + n, acc[j][r]);
    }
  }
}

// lat = lfla1 @ t (WMMA) fused with:
// f_out = C*(z - fa0*f_in + fa0*mean + lat) + (1-C)*hnow   (z already in f_out)
__global__ void k_wmma_combine(const float* __restrict__ lfla1, const float* __restrict__ t,
                               const float* __restrict__ f_in, const float* __restrict__ hnow,
                               const float* __restrict__ fa0, const float* __restrict__ sumf,
                               float* __restrict__ f_out) {
  int lane = threadIdx.x & 31;
  int wv = (blockIdx.x * blockDim.x + threadIdx.x) >> 5;
  int nw = (gridDim.x * blockDim.x) >> 5;
  int l15 = lane & 15;
  int hi = lane >> 4;
  const float invn = 1.0f / (float)NU;
  const int nrows = NU / 16;  // 6250, exact
  for (int row = wv; row < nrows; row += nw) {
    int u0 = row * 16;
    for (int tk = 0; tk < 8; ++tk) {
      v8f acc = {};
#pragma unroll
      for (int k4 = 0; k4 < GD; k4 += 4) {
        v2f a, b;
        // A[M=user_local, K=d] = lfla1[u0+M, k4+K]
        a.x = lfla1[(u0 + l15) * GD + k4 + 2 * hi + 0];
        a.y = lfla1[(u0 + l15) * GD + k4 + 2 * hi + 1];
        // B[K=d, N=kcol] = t[k4+K, tk*16+N]
        b.x = t[(k4 + 2 * hi + 0) * KD + tk * 16 + l15];
        b.y = t[(k4 + 2 * hi + 1) * KD + tk * 16 + l15];
        acc = __builtin_amdgcn_wmma_f32_16x16x4_f32(false, a, false, b, (short)0, acc,
                                                    false, false);
      }
#pragma unroll
      for (int r = 0; r < 8; ++r) {
        int u = u0 + hi * 8 + r;
        int kc = tk * 16 + l15;
        size_t idx = (size_t)u * KD + kc;
        float z = f_out[idx];
        float fa = fa0[u];
        float mf = sumf[kc] * invn;
        f_out[idx] = CMIX * (z - fa * f_in[idx] + fa * mf + acc[r]) + (1.0f - CMIX) * hnow[idx];
      }
    }
  }
}

__global__ void k_add_eps(const float* __restrict__ src, float* __restrict__ dst, int n) {
  int i = blockIdx.x * blockDim.x + threadIdx.x;
  int st = gridDim.x * blockDim.x;
  for (; i < n; i += st) dst[i] = src[i] + EPSV;
}

extern "C" void kernel_launch(void* const* d_in, const int* in_sizes, int n_in,
                              void* d_out, int out_size, void* d_ws, size_t ws_size,
                              hipStream_t stream) {
  const float* flp  = (const float*)d_in[0];
  const float* hnow = (const float*)d_in[1];
  const float* w1   = (const float*)d_in[2];
  const float* w2   = (const float*)d_in[3];
  const float* la1  = (const float*)d_in[4];
  const float* la2  = (const float*)d_in[5];
  const int*   r1   = (const int*)d_in[6];
  const int*   r2   = (const int*)d_in[7];

  float* out = (float*)d_out;
  float* o_lfw1  = out;
  float* o_lfw2  = out + (size_t)NE;
  float* o_lfla1 = out + 2 * (size_t)NE;
  float* o_lfla2 = o_lfla1 + (size_t)NU * GD;
  float* o_flp   = o_lfla2 + (size_t)NU * GD;

  float* ws = (float*)d_ws;
  float* ws_f     = ws;  ws += (size_t)NU * KD;   // 12.8M
  float* ws_y     = ws;  ws += (size_t)NI * KD;   //  6.4M
  float* ws_sumw1 = ws;  ws += NU;
  float* ws_sumw2 = ws;  ws += NI;
  float* ws_fa0   = ws;  ws += NU;
  float* ws_col2  = ws;  ws += GD;
  float* ws_t     = ws;  ws += GD * KD;           // 8192 (contiguous with sumf)
  float* ws_sumf  = ws;  ws += KD;                // 128

  // ---- normalization phase ----
  k_user_init<<<(NU + 255) / 256, 256, 0, stream>>>(la1, o_lfla1, ws_sumw1);
  k_small_init<<<(NU + 255) / 256, 256, 0, stream>>>(ws_sumw2, ws_fa0, ws_col2);
  k_la2_exp_colsum<<<256, 256, 0, stream>>>(la2, o_lfla2, ws_col2);
  k_edge_exp<<<2048, 256, 0, stream>>>(w1, w2, r1, r2, o_lfw1, o_lfw2, ws_sumw1, ws_sumw2);
  k_edge_norm<<<2048, 256, 0, stream>>>(r1, r2, o_lfw1, o_lfw2, ws_sumw1, ws_sumw2, ws_fa0);
  k_la_norm<<<(NU * GD + 255) / 256, 256, 0, stream>>>(o_lfla1, o_lfla2, ws_sumw1, ws_col2);

  // ---- propagation: ping-pong ws_f <-> o_flp, final lands in ws_f ----
  const float* fin = flp;
  float* outs[DEPTH] = {ws_f, o_flp, ws_f};
  for (int it = 0; it < DEPTH; ++it) {
    float* fout = outs[it];
    k_zero<<<2048, 256, 0, stream>>>(ws_y, NI * KD);
    k_zero<<<2048, 256, 0, stream>>>(fout, NU * KD);
    k_zero<<<33, 256, 0, stream>>>(ws_t, GD * KD + KD);  // t + sumf (adjacent)
    k_colsum_f<<<256, 256, 0, stream>>>(fin, ws_sumf);
    k_scatter_y<<<4096, 256, 0, stream>>>(fin, o_lfw2, r1, r2, ws_y);
    k_scatter_z<<<4096, 256, 0, stream>>>(ws_y, o_lfw1, r1, r2, fout);
    k_wmma_t<<<250, 256, 0, stream>>>(o_lfla2, fin, ws_t);
    k_wmma_combine<<<800, 256, 0, stream>>>(o_lfla1, ws_t, fin, hnow, ws_fa0, ws_sumf, fout);
    fin = fout;
  }
  k_add_eps<<<2048, 256, 0, stream>>>(ws_f, o_flp, NU * KD);
}